// WeightedHausdorffDistance_12017318494397
// MI455X (gfx1250) — compile-verified
//
#include <hip/hip_runtime.h>
#include <math.h>

typedef __attribute__((ext_vector_type(2))) float v2f;
typedef __attribute__((ext_vector_type(4))) float v4f;
typedef __attribute__((ext_vector_type(8))) float v8f;

#define HH 512
#define WW 512
#define NPTS 256
#define NGROUP ((HH * WW) / 16)   // 16384 groups of 16 pixels (one grid row chunk)

// Plain v_min_num_f32 via inline asm: opaque to LLVM, so no canonicalize ops
// (v_max v,v,v) get inserted around it and it cannot be pattern-rewritten.
__device__ __forceinline__ float vminf_fast(float a, float b) {
    float r;
    asm("v_min_num_f32 %0, %1, %2" : "=v"(r) : "v"(a), "v"(b));
    return r;
}

// DPP lane permute within row of 16 (full-rate VALU, no DS path)
#define DPP_CTRL_XOR1  0xB1   // quad_perm(1,0,3,2)
#define DPP_CTRL_XOR2  0x4E   // quad_perm(2,3,0,1)
#define DPP_CTRL_HMIRR 0x141  // ROW_HALF_MIRROR (lane ^ 7 within 8)
#define DPP_CTRL_MIRR  0x140  // ROW_MIRROR     (lane ^ 15 within 16)

#define DPP_MIN_STEP(v, ctrl)                                              \
    do {                                                                   \
        int _xi = __builtin_bit_cast(int, (v));                            \
        int _ri = __builtin_amdgcn_mov_dpp(_xi, (ctrl), 0xF, 0xF, true);   \
        (v) = vminf_fast((v), __builtin_bit_cast(float, _ri));             \
    } while (0)

// ds_swizzle group-of-32 xor patterns (ISA §11): offset = (xor<<10) | 0x1f.
// ENC must be a constant expression -> template parameter.
template <int ENC>
__device__ __forceinline__ float swz_f(float x) {
    int r = __builtin_amdgcn_ds_swizzle(__builtin_bit_cast(int, x), ENC);
    return __builtin_bit_cast(float, r);
}
#define SWZ_XOR(mask) (((mask) << 10) | 0x1F)

// ---------------------------------------------------------------------------
// Kernel 0: global max distance. dist(g) is convex in grid coords, so the max
// over the lattice is attained at one of the 4 corners -> 256 pts x 4 corners.
// ---------------------------------------------------------------------------
__global__ __launch_bounds__(256) void k_maxdist(const float* __restrict__ pts,
                                                 float* __restrict__ g_max) {
    int n = threadIdx.x;                       // 0..255 == point id
    float py = pts[2 * n + 0];
    float px = pts[2 * n + 1];
    float m = 0.0f;
#pragma unroll
    for (int cy = 0; cy < 2; ++cy) {
#pragma unroll
        for (int cx = 0; cx < 2; ++cx) {
            float dy = ((cy ? 511.0f : 0.0f) - py) * (1.0f / 512.0f);
            float dx = ((cx ? 511.0f : 0.0f) - px) * (1.0f / 512.0f);
            m = fmaxf(m, __builtin_amdgcn_sqrtf(dy * dy + dx * dx));
        }
    }
    __shared__ float s[256];
    s[n] = m;
    __syncthreads();
    for (int off = 128; off > 0; off >>= 1) {
        if (n < off) s[n] = fmaxf(s[n], s[n + off]);
        __syncthreads();
    }
    if (n == 0) g_max[0] = s[0];
}

// ---------------------------------------------------------------------------
// Main kernel. Each wave owns 16-pixel groups (grid-stride). For each group it
// runs 16 WMMA f32 16x16x4 ops (one per 16-point tile):
//   A row m  = (-2*gy/H^2, -2*gx/W^2, u_m, 1)   u_m = (gy/H)^2+(gx/W)^2
//   B col n  = (py, px, 1, v_n)                 v_n = (py/H)^2+(px/W)^2
//   D[m][n]  = squared normalized distance.
// Per element: v_min (asm), v_sqrt with free |.| modifier, fma, rcp, add —
// 2 transcendentals/element is the floor for this op.
// Term1: min over points kept on d^2, DPP butterfly-min across the 16-lane row.
// Term2: r = 1/(hm*d + (1-hm)*maxd); half-row sums joined with ds_swizzle x16.
// ---------------------------------------------------------------------------
__global__ __launch_bounds__(256) void k_main(const float* __restrict__ hm,
                                              const float* __restrict__ pts,
                                              const float* __restrict__ g_max,
                                              float* __restrict__ pacc_part,
                                              float* __restrict__ t1_part,
                                              float* __restrict__ habs_part,
                                              int nblk) {
    const int lane = threadIdx.x & 31;
    const int wv   = threadIdx.x >> 5;          // wave in block: 0..7
    const int j    = lane & 15;                 // column / pixel-in-group index
    const int half = lane >> 4;                 // 0: K0/K1+M0..7, 1: K2/K3+M8..15
    const float maxd  = g_max[0];
    const float invS  = 1.0f / 512.0f;
    const float invS2 = invS * invS;

    // B fragments for all 16 point tiles (invariant over pixel groups).
    v2f bfrag[16];
#pragma unroll
    for (int t = 0; t < 16; ++t) {
        int n = t * 16 + j;
        float py = pts[2 * n + 0];
        float px = pts[2 * n + 1];
        if (half == 0) {                         // K=0,1 rows: (py, px)
            bfrag[t].x = py;
            bfrag[t].y = px;
        } else {                                 // K=2,3 rows: (1, v_n)
            float sy = py * invS, sx = px * invS;
            bfrag[t].x = 1.0f;
            bfrag[t].y = sy * sy + sx * sx;
        }
    }

    float acc[16];                               // per-point-tile column sums
#pragma unroll
    for (int t = 0; t < 16; ++t) acc[t] = 0.0f;
    float t1_acc = 0.0f, habs_acc = 0.0f;

    const int waves_total = nblk * 8;
    const int wave_id     = blockIdx.x * 8 + wv;

    for (int g = wave_id; g < NGROUP; g += waves_total) {
        const int base = g * 16;                 // 16 consecutive pixels, same row
        const float gy = (float)(base >> 9);
        const float gx = (float)((base & 511) + j);

        v2f afrag;
        if (half == 0) {                         // K=0,1: (-2gy/H^2, -2gx/W^2)
            afrag.x = -2.0f * gy * invS2;
            afrag.y = -2.0f * gx * invS2;
        } else {                                 // K=2,3: (u_m, 1)
            float sy = gy * invS, sx = gx * invS;
            afrag.x = sy * sy + sx * sx;
            afrag.y = 1.0f;
        }

        // hm for the 8 M-rows this half owns: 8 consecutive floats -> 2x b128
        // broadcast loads (all 16 lanes of a half read the same addresses).
        const v4f* hp = (const v4f*)(hm + base + 8 * half);
        v4f h0 = hp[0];
        v4f h1 = hp[1];
        float a_r[8], c_r[8];
#pragma unroll
        for (int i = 0; i < 8; ++i) {
            a_r[i] = (i < 4) ? h0[i] : h1[i - 4];
            c_r[i] = (1.0f - a_r[i]) * maxd;
        }
        if (j < 8) habs_acc += fabsf(a_r[j]);    // lanes 0-7 -> M0-7, 16-23 -> M8-15

        float dmin2[8];
#pragma unroll
        for (int i = 0; i < 8; ++i) dmin2[i] = 3.4e38f;

#pragma unroll
        for (int t = 0; t < 16; ++t) {
            v8f cz = {};
            v8f d2 = __builtin_amdgcn_wmma_f32_16x16x4_f32(
                false, afrag, false, bfrag[t], (short)0, cz, false, false);
            float rs = 0.0f;
#pragma unroll
            for (int i = 0; i < 8; ++i) {
                float q = d2[i];
                dmin2[i] = vminf_fast(dmin2[i], q);
                // |q| guards -eps from cancellation; fabs folds into the
                // v_sqrt source modifier (zero extra instructions).
                float dd = __builtin_amdgcn_sqrtf(fabsf(q));
                float wd = fmaf(a_r[i], dd, c_r[i]);
                rs += __builtin_amdgcn_rcpf(wd);
            }
            acc[t] += rs + swz_f<SWZ_XOR(0x10)>(rs);  // add other half's rows
        }

        // Term 1: min over the 16 columns of each row — DPP butterfly.
#pragma unroll
        for (int i = 0; i < 8; ++i) {
            float v = dmin2[i];
            DPP_MIN_STEP(v, DPP_CTRL_XOR1);
            DPP_MIN_STEP(v, DPP_CTRL_XOR2);
            DPP_MIN_STEP(v, DPP_CTRL_HMIRR);
            DPP_MIN_STEP(v, DPP_CTRL_MIRR);
            dmin2[i] = v;
        }
        if (j < 8)                               // each pixel contributed once
            t1_acc += a_r[j] * __builtin_amdgcn_sqrtf(fabsf(dmin2[j]));
    }

    // ---- deterministic block reduction (no atomics) ----
    __shared__ float s_p[8][256];
    __shared__ float s_t1[8];
    __shared__ float s_hb[8];
    if (lane < 16) {
#pragma unroll
        for (int t = 0; t < 16; ++t) s_p[wv][t * 16 + j] = acc[t];
    }
    float tw = t1_acc, hw = habs_acc;
    tw += swz_f<SWZ_XOR(1)>(tw);
    hw += swz_f<SWZ_XOR(1)>(hw);
    tw += swz_f<SWZ_XOR(2)>(tw);
    hw += swz_f<SWZ_XOR(2)>(hw);
    tw += swz_f<SWZ_XOR(4)>(tw);
    hw += swz_f<SWZ_XOR(4)>(hw);
    tw += swz_f<SWZ_XOR(8)>(tw);
    hw += swz_f<SWZ_XOR(8)>(hw);
    tw += swz_f<SWZ_XOR(16)>(tw);
    hw += swz_f<SWZ_XOR(16)>(hw);
    if (lane == 0) { s_t1[wv] = tw; s_hb[wv] = hw; }
    __syncthreads();

    const int tid = threadIdx.x;
    float ps = 0.0f;
#pragma unroll
    for (int w = 0; w < 8; ++w) ps += s_p[w][tid];
    pacc_part[blockIdx.x * 256 + tid] = ps;
    if (tid == 0) {
        float a = 0.0f, b = 0.0f;
#pragma unroll
        for (int w = 0; w < 8; ++w) { a += s_t1[w]; b += s_hb[w]; }
        t1_part[blockIdx.x]   = a;
        habs_part[blockIdx.x] = b;
    }
}

// ---------------------------------------------------------------------------
// Finalize: combine per-block partials, apply power-mean (alpha=-1) and the
// heatmap normalization, write the scalar result.
// ---------------------------------------------------------------------------
__global__ __launch_bounds__(256) void k_final(const float* __restrict__ pacc_part,
                                               const float* __restrict__ t1_part,
                                               const float* __restrict__ habs_part,
                                               int nblk, float* __restrict__ out) {
    int n = threadIdx.x;                         // point id 0..255
    float s = 0.0f;
    for (int b = 0; b < nblk; ++b) s += pacc_part[b * 256 + n];
    // soft_min[n] = (mean wd^-1)^-1 = (H*W) / sum(wd^-1)
    float soft = (512.0f * 512.0f) / s;
    __shared__ float sh[256];
    sh[n] = soft;
    __syncthreads();
    for (int off = 128; off > 0; off >>= 1) {
        if (n < off) sh[n] += sh[n + off];
        __syncthreads();
    }
    if (n == 0) {
        float t1 = 0.0f, hb = 0.0f;
        for (int b = 0; b < nblk; ++b) { t1 += t1_part[b]; hb += habs_part[b]; }
        float weighted_min = t1 / (1e-5f + hb);
        out[0] = weighted_min + sh[0] * (1.0f / 256.0f);
    }
}

// ---------------------------------------------------------------------------
extern "C" void kernel_launch(void* const* d_in, const int* in_sizes, int n_in,
                              void* d_out, int out_size, void* d_ws, size_t ws_size,
                              hipStream_t stream) {
    const float* hm  = (const float*)d_in[0];   // [512*512] f32
    const float* pts = (const float*)d_in[1];   // [256*2] f32 (y, x)
    float* out = (float*)d_out;                 // scalar f32

    float* f = (float*)d_ws;
    // ws layout (floats): [0]=maxd | +64: t1_part | +1088: habs_part | +2112: pacc
    float* g_max     = f;
    float* t1_part   = f + 64;
    float* habs_part = f + 64 + 1024;
    float* pacc_part = f + 64 + 2048;

    int nblk = 512;
    long availF = (long)(ws_size / 4);
    long need   = 64 + 2048 + (long)nblk * 256;
    if (availF < need) {
        long nb = (availF - 2112) / 256;
        if (nb < 1) nb = 1;
        if (nb > 512) nb = 512;
        nblk = (int)nb;
    }

    k_maxdist<<<1, 256, 0, stream>>>(pts, g_max);
    k_main<<<nblk, 256, 0, stream>>>(hm, pts, g_max, pacc_part, t1_part, habs_part, nblk);
    k_final<<<1, 256, 0, stream>>>(pacc_part, t1_part, habs_part, nblk, out);
}